// DWAMiddleLayer_23794118819889
// MI455X (gfx1250) — compile-verified
//
#include <hip/hip_runtime.h>

typedef float v2f __attribute__((ext_vector_type(2)));
typedef float v8f __attribute__((ext_vector_type(8)));

#define D_A      1024
#define D_B      1024
#define R_RANK   16
#define N_POOL   64
#define BATCH    256
#define POOL_DIM (D_B * R_RANK + R_RANK * D_A + D_B) /* 33792 */
#define U_END    (D_B * R_RANK)                      /* 16384 */
#define V_END    (U_END + R_RANK * D_A)              /* 32768 */
#define LN_EPS   1e-5f

// ---------------------------------------------------------------------------
// Kernel 1: s[b, k] = alpha[b, k/16] * sum_a h_A[b,a] * V[k/16, k%16, a]
//   GEMM M=256(b) N=1024(k=n*16+r) K=1024(a), alpha scale fused in epilogue.
//   Each wave: 16x64 output strip (4 N-tiles), fp32 WMMA 16x16x4.
// ---------------------------------------------------------------------------
__global__ __launch_bounds__(128)
void k1_s_gemm(const float* __restrict__ hA,
               const float* __restrict__ pool,
               const float* __restrict__ alpha,
               float* __restrict__ s_out) {
  const int lane  = threadIdx.x & 31;
  const int wave  = threadIdx.x >> 5;
  const int gw    = blockIdx.x * 4 + wave;   // 0..255
  const int tm    = gw >> 4;                 // row tile 0..15
  const int strip = gw & 15;                 // 64-col strip 0..15
  const int row0  = tm * 16;
  const int l15   = lane & 15;
  const int half  = lane >> 4;               // 0 or 1
  const int kk    = half << 1;               // K sub-offset 0 or 2

  v8f acc[4] = {};
  const float* Arow = hA + (size_t)(row0 + l15) * D_A + kk;
  const float* Bp[4];
#pragma unroll
  for (int t = 0; t < 4; ++t) {
    const int n = strip * 4 + t;             // pool row index
    Bp[t] = pool + (size_t)n * POOL_DIM + U_END + (size_t)l15 * D_A + kk;
  }

  for (int k0 = 0; k0 < D_A; k0 += 4) {
    v2f a = *(const v2f*)(Arow + k0);
#pragma unroll
    for (int t = 0; t < 4; ++t) {
      v2f b = *(const v2f*)(Bp[t] + k0);
      acc[t] = __builtin_amdgcn_wmma_f32_16x16x4_f32(
          false, a, false, b, (short)0, acc[t], false, false);
    }
  }

#pragma unroll
  for (int t = 0; t < 4; ++t) {
    const int n   = strip * 4 + t;
    const int col = n * 16 + l15;            // k index
#pragma unroll
    for (int v = 0; v < 8; ++v) {
      const int brow = row0 + v + (half << 3);
      s_out[(size_t)brow * (R_RANK * N_POOL) + col] =
          acc[t][v] * alpha[(size_t)brow * N_POOL + n];
    }
  }
}

// ---------------------------------------------------------------------------
// Kernel 2: pre[b,c] = h_A[b,c] + gamma * ( h_A@W_base^T + s@U_flat
//                                           + alpha@bias_pool + b_base[c] )
//   Three K-loops accumulate into the same v8f[4] tile accumulators.
// ---------------------------------------------------------------------------
__global__ __launch_bounds__(128)
void k2_fused_gemm(const float* __restrict__ hA,
                   const float* __restrict__ pool,
                   const float* __restrict__ alpha,
                   const float* __restrict__ Wb,
                   const float* __restrict__ bb,
                   const float* __restrict__ gamma_p,
                   const float* __restrict__ s_in,
                   float* __restrict__ pre) {
  const int lane  = threadIdx.x & 31;
  const int wave  = threadIdx.x >> 5;
  const int gw    = blockIdx.x * 4 + wave;   // 0..255
  const int tm    = gw >> 4;
  const int strip = gw & 15;
  const int row0  = tm * 16;
  const int col0  = strip * 64;
  const int l15   = lane & 15;
  const int half  = lane >> 4;
  const int kk    = half << 1;

  v8f acc[4] = {};

  // --- Part 1: base = h_A @ W_base^T.  B[a][c] = Wb[c*D_A + a]
  {
    const float* Arow = hA + (size_t)(row0 + l15) * D_A + kk;
    const float* Bp[4];
#pragma unroll
    for (int t = 0; t < 4; ++t)
      Bp[t] = Wb + (size_t)(col0 + t * 16 + l15) * D_A + kk;
    for (int k0 = 0; k0 < D_A; k0 += 4) {
      v2f a = *(const v2f*)(Arow + k0);
#pragma unroll
      for (int t = 0; t < 4; ++t) {
        v2f b = *(const v2f*)(Bp[t] + k0);
        acc[t] = __builtin_amdgcn_wmma_f32_16x16x4_f32(
            false, a, false, b, (short)0, acc[t], false, false);
      }
    }
  }

  // --- Part 2: delta = s @ U_flat.  B[k][c] = pool[(k>>4)*POOL + c*16 + (k&15)]
  {
    const float* Srow = s_in + (size_t)(row0 + l15) * (R_RANK * N_POOL) + kk;
    for (int k0 = 0; k0 < R_RANK * N_POOL; k0 += 4) {
      v2f a = *(const v2f*)(Srow + k0);
      const int kidx = k0 + kk;
      const int n    = kidx >> 4;
      const int r    = kidx & 15;            // even, r+1 stays in-block
      const float* base = pool + (size_t)n * POOL_DIM + r;
#pragma unroll
      for (int t = 0; t < 4; ++t) {
        const int c = col0 + t * 16 + l15;
        v2f b = *(const v2f*)(base + (size_t)c * R_RANK);
        acc[t] = __builtin_amdgcn_wmma_f32_16x16x4_f32(
            false, a, false, b, (short)0, acc[t], false, false);
      }
    }
  }

  // --- Part 3: bias_delta = alpha @ bias_pool.  B[n][c] = pool[n*POOL + V_END + c]
  {
    const float* Alrow = alpha + (size_t)(row0 + l15) * N_POOL + kk;
    for (int k0 = 0; k0 < N_POOL; k0 += 4) {
      v2f a = *(const v2f*)(Alrow + k0);
      const int n0 = k0 + kk;
      const float* p0 = pool + (size_t)n0 * POOL_DIM + V_END;
      const float* p1 = p0 + POOL_DIM;
#pragma unroll
      for (int t = 0; t < 4; ++t) {
        const int c = col0 + t * 16 + l15;
        v2f b;
        b.x = p0[c];
        b.y = p1[c];
        acc[t] = __builtin_amdgcn_wmma_f32_16x16x4_f32(
            false, a, false, b, (short)0, acc[t], false, false);
      }
    }
  }

  // --- Epilogue: residual + gamma scale
  const float g = gamma_p[0];
#pragma unroll
  for (int t = 0; t < 4; ++t) {
    const int c   = col0 + t * 16 + l15;
    const float bbc = bb[c];
#pragma unroll
    for (int v = 0; v < 8; ++v) {
      const int brow = row0 + v + (half << 3);
      const float ht = acc[t][v] + bbc;
      pre[(size_t)brow * D_A + c] = hA[(size_t)brow * D_A + c] + g * ht;
    }
  }
}

// ---------------------------------------------------------------------------
// Kernel 3: row-wise LayerNorm, one 256-thread block (8 waves) per batch row.
// ---------------------------------------------------------------------------
__global__ __launch_bounds__(256)
void k3_layernorm(const float* __restrict__ pre,
                  const float* __restrict__ ln_scale,
                  const float* __restrict__ ln_bias,
                  float* __restrict__ out) {
  __shared__ float red[256];
  const int b   = blockIdx.x;
  const int tid = threadIdx.x;
  const float* x = pre + (size_t)b * D_A;

  float vals[4];
  float ssum = 0.f;
#pragma unroll
  for (int i = 0; i < 4; ++i) {
    vals[i] = x[tid + i * 256];
    ssum += vals[i];
  }
  red[tid] = ssum;
  __syncthreads();
  for (int off = 128; off > 0; off >>= 1) {
    if (tid < off) red[tid] += red[tid + off];
    __syncthreads();
  }
  const float mean = red[0] * (1.0f / D_A);
  __syncthreads();

  float sq = 0.f;
#pragma unroll
  for (int i = 0; i < 4; ++i) {
    const float d = vals[i] - mean;
    sq += d * d;
  }
  red[tid] = sq;
  __syncthreads();
  for (int off = 128; off > 0; off >>= 1) {
    if (tid < off) red[tid] += red[tid + off];
    __syncthreads();
  }
  const float var = red[0] * (1.0f / D_A);
  const float inv = rsqrtf(var + LN_EPS);

#pragma unroll
  for (int i = 0; i < 4; ++i) {
    const int c = tid + i * 256;
    out[(size_t)b * D_A + c] = (vals[i] - mean) * inv * ln_scale[c] + ln_bias[c];
  }
}

// ---------------------------------------------------------------------------
extern "C" void kernel_launch(void* const* d_in, const int* in_sizes, int n_in,
                              void* d_out, int out_size, void* d_ws, size_t ws_size,
                              hipStream_t stream) {
  const float* hA       = (const float*)d_in[0];  // (256, 1024)
  const float* pool     = (const float*)d_in[1];  // (64, 33792)
  const float* alpha    = (const float*)d_in[2];  // (256, 64)
  const float* Wb       = (const float*)d_in[3];  // (1024, 1024)
  const float* bb       = (const float*)d_in[4];  // (1024,)
  const float* gamma_p  = (const float*)d_in[5];  // scalar
  const float* ln_scale = (const float*)d_in[6];  // (1024,)
  const float* ln_bias  = (const float*)d_in[7];  // (1024,)
  float* out = (float*)d_out;

  float* s_ws   = (float*)d_ws;                    // 256*1024 f32 = 1 MB
  float* pre_ws = s_ws + (size_t)BATCH * (R_RANK * N_POOL); // 1 MB more

  // 256 waves total per GEMM kernel, 4 waves (128 thr) per block -> 64 blocks
  k1_s_gemm<<<64, 128, 0, stream>>>(hA, pool, alpha, s_ws);
  k2_fused_gemm<<<64, 128, 0, stream>>>(hA, pool, alpha, Wb, bb, gamma_p,
                                        s_ws, pre_ws);
  k3_layernorm<<<BATCH, 256, 0, stream>>>(pre_ws, ln_scale, ln_bias, out);
}